// GlowTTS_35974646071700
// MI455X (gfx1250) — compile-verified
//
#include <hip/hip_runtime.h>
#include <math.h>

typedef __attribute__((ext_vector_type(2))) float v2f;
typedef __attribute__((ext_vector_type(8))) float v8f;

#define B_  16
#define C_  80
#define TX_ 512
#define TY_ 2048
#define K_  160
#define NEGV -1000000000.0f

// ---------------------------------------------------------------------------
// Pack A operand: Apack[b][t][k] = { o_scale (k<80) , o_mean*o_scale (k>=80) }
// and per-(b,t) scalar S1 = logp1 + logp4.
// ---------------------------------------------------------------------------
__global__ __launch_bounds__(256) void packA_kernel(
    const float* __restrict__ o_mean, const float* __restrict__ o_ls,
    float* __restrict__ Apack, float* __restrict__ S1)
{
    __shared__ float red[8][32];
    int b = blockIdx.y;
    int t = blockIdx.x * 32 + threadIdx.x;
    float partial = 0.f;
    for (int c = threadIdx.y; c < C_; c += 8) {
        float ls = o_ls  [(size_t)(b*C_ + c)*TX_ + t];
        float m  = o_mean[(size_t)(b*C_ + c)*TX_ + t];
        float sc = expf(-2.f * ls);
        size_t row = (size_t)(b*TX_ + t) * K_;
        Apack[row + c]       = sc;
        Apack[row + C_ + c]  = m * sc;
        partial += (-0.91893853320467274f - ls) - 0.5f * m * m * sc;
    }
    red[threadIdx.y][threadIdx.x] = partial;
    __syncthreads();
    if (threadIdx.y == 0) {
        float s = 0.f;
        #pragma unroll
        for (int i = 0; i < 8; ++i) s += red[i][threadIdx.x];
        S1[b*TX_ + t] = s;
    }
}

// ---------------------------------------------------------------------------
// Pack B operand: Bpack[b][s][k] = { -0.5*z^2 (k<80) , z (k>=80) } with y-mask
// applied; also emits output 0 (z transposed, masked) on the way.
// ---------------------------------------------------------------------------
__global__ __launch_bounds__(256) void packB_kernel(
    const float* __restrict__ z, const int* __restrict__ yl,
    float* __restrict__ Bpack, float* __restrict__ out0)
{
    int b = blockIdx.y;
    int s = blockIdx.x * 32 + threadIdx.x;
    float ym = (s < yl[b]) ? 1.f : 0.f;
    for (int c = threadIdx.y; c < C_; c += 8) {
        float zv = z[(size_t)(b*C_ + c)*TY_ + s] * ym;
        size_t row = (size_t)(b*TY_ + s) * K_;
        Bpack[row + c]      = -0.5f * zv * zv;
        Bpack[row + C_ + c] = zv;
        out0[(size_t)(b*TY_ + s)*C_ + c] = zv;
    }
}

// ---------------------------------------------------------------------------
// logp GEMM via V_WMMA_F32_16X16X4_F32. One wave per 16x16 (t,s) tile,
// K=160 in 40 steps of 4. Epilogue: S1 preloaded unconditionally as two
// float4 loads (8 contiguous floats per lane), then branch-free masked
// selects; tile transposed through LDS and stored s-major to VAL[b][s][t].
// ---------------------------------------------------------------------------
__global__ __launch_bounds__(256) void logp_wmma_kernel(
    const float* __restrict__ Apack, const float* __restrict__ Bpack,
    const float* __restrict__ S1, const int* __restrict__ xl,
    const int* __restrict__ yl, float* __restrict__ VAL)
{
    __shared__ float tile[8][16][17];
    int lane = threadIdx.x & 31;
    int wv   = threadIdx.x >> 5;
    int b    = blockIdx.z;
    int t0   = blockIdx.y * 16;
    int s0   = (blockIdx.x * 8 + wv) * 16;
    int rlane = lane & 15;
    int koff  = (lane >> 4) << 1;

    const float* Ar = Apack + (size_t)(b*TX_ + t0 + rlane)*K_ + koff;
    const float* Br = Bpack + (size_t)(b*TY_ + s0 + rlane)*K_ + koff;

    v8f acc = {0.f,0.f,0.f,0.f,0.f,0.f,0.f,0.f};
    #pragma unroll 4
    for (int k = 0; k < K_; k += 4) {
        float2 af = *(const float2*)(Ar + k);
        float2 bf = *(const float2*)(Br + k);
        v2f a;  a.x  = af.x; a.y  = af.y;
        v2f bb; bb.x = bf.x; bb.y = bf.y;
        acc = __builtin_amdgcn_wmma_f32_16x16x4_f32(
            false, a, false, bb, (short)0, acc, false, false);
    }

    int txl = xl[b], tyl = yl[b];
    int off = (lane >> 4) << 3;   // row offset 0 or 8
    int n   = rlane;              // N (s) index
    int s   = s0 + n;

    // Unconditional preload of the 8 contiguous S1 values this lane needs.
    const float* s1p = S1 + b*TX_ + t0 + off;
    float4 s1a = *(const float4*)(s1p);
    float4 s1b = *(const float4*)(s1p + 4);
    float s1v[8] = { s1a.x, s1a.y, s1a.z, s1a.w, s1b.x, s1b.y, s1b.z, s1b.w };

    bool sm = (s < tyl);
    #pragma unroll
    for (int r = 0; r < 8; ++r) {
        int t = t0 + r + off;
        float v = acc[r] + s1v[r];
        v = (sm && (t < txl)) ? v : 0.f;    // branch-free: no loads guarded
        tile[wv][r + off][n] = v;
    }
    __syncthreads();
    int j  = lane >> 1;           // s row within tile
    int ms = (lane & 1) << 3;     // t half
    float4 w0, w1;
    w0.x = tile[wv][ms+0][j]; w0.y = tile[wv][ms+1][j];
    w0.z = tile[wv][ms+2][j]; w0.w = tile[wv][ms+3][j];
    w1.x = tile[wv][ms+4][j]; w1.y = tile[wv][ms+5][j];
    w1.z = tile[wv][ms+6][j]; w1.w = tile[wv][ms+7][j];
    float* dst = VAL + (size_t)(b*TY_ + s0 + j)*TX_ + t0 + ms;
    *(float4*)dst       = w0;
    *(float4*)(dst + 4) = w1;
}

// ---------------------------------------------------------------------------
// MAS forward: one block per batch, 512 threads, sequential over Ty columns.
// Running V column lives in LDS (double-buffered). Instead of storing V,
// store one decision bit per (x,y): V[x-1][y] > V[x][y], packed via wave32
// ballot -> 2 MB total.
// ---------------------------------------------------------------------------
__global__ __launch_bounds__(512) void mas_forward_kernel(
    const float* __restrict__ VAL, unsigned* __restrict__ DEC)
{
    __shared__ float vbuf[2][TX_];
    int b = blockIdx.x;
    int t = threadIdx.x;
    int lane = t & 31, wv = t >> 5;
    const float* valb = VAL + (size_t)b*TY_*TX_;
    unsigned*    decb = DEC + (size_t)b*TY_*16;

    vbuf[0][t] = (t == 0) ? valb[0] : NEGV;
    __syncthreads();
    int p = 0;
    for (int y = 1; y < TY_; ++y) {
        float vc  = vbuf[p][t];
        float vm1 = (t > 0) ? vbuf[p][t-1] : NEGV;
        unsigned m = __builtin_amdgcn_ballot_w32((t > 0) && (vm1 > vc));
        if (lane == 0) decb[(size_t)(y-1)*16 + wv] = m;
        float col = valb[(size_t)y*TX_ + t];
        vbuf[p ^ 1][t] = col + fmaxf(vc, vm1);
        __syncthreads();
        p ^= 1;
    }
}

// ---------------------------------------------------------------------------
// MAS backward: one block per batch. Decision bits staged through LDS in
// 512-column chunks (32 KB) so the serial backtrack chases LDS, not HBM.
// Emits IDX[b][y] (chosen t per y, -1 if inactive), durations histogram and
// output 4 (log1p(dur)*x_mask).
// ---------------------------------------------------------------------------
__global__ __launch_bounds__(256) void mas_backward_kernel(
    const unsigned* __restrict__ DEC, const int* __restrict__ xl,
    const int* __restrict__ yl, int* __restrict__ IDX, float* __restrict__ out4)
{
    __shared__ unsigned decS[512*16];
    __shared__ int      idxS[TY_];
    __shared__ unsigned durS[TX_];
    __shared__ int      carry;
    int b = blockIdx.x;
    int tid = threadIdx.x;
    int txl = xl[b], tyl = yl[b];
    if (tid == 0) carry = txl - 1;
    for (int i = tid; i < TX_; i += 256) durS[i] = 0;
    const unsigned* decb = DEC + (size_t)b*TY_*16;

    for (int chunk = 3; chunk >= 0; --chunk) {
        int cb = chunk * 512;
        int lo = (cb == 0) ? 0 : cb - 1;
        __syncthreads();
        for (int i = tid; i < 512*16; i += 256)
            decS[i] = decb[(size_t)lo*16 + i];
        __syncthreads();
        if (tid == 0) {
            int idx = carry;
            for (int y = cb + 511; y >= cb; --y) {
                if (y == tyl - 1) idx = txl - 1;
                bool active = y < tyl;
                idxS[y] = active ? idx : -1;
                int yc = (y > 0) ? (y - 1) : 0;
                unsigned mw = decS[(yc - lo)*16 + (idx >> 5)];
                bool dec  = (mw >> (idx & 31)) & 1u;
                bool move = active && (idx > 0) && ((idx == y) || dec);
                idx -= (int)move;
            }
            carry = idx;
        }
    }
    __syncthreads();
    for (int y = tid; y < TY_; y += 256) {
        int id = idxS[y];
        IDX[b*TY_ + y] = id;
        if (id >= 0) atomicAdd(&durS[id], 1u);
    }
    __syncthreads();
    for (int t = tid; t < TX_; t += 256) {
        float d = (float)durS[t];
        out4[b*TX_ + t] = (t < txl) ? log1pf(d) : 0.f;
    }
}

// ---------------------------------------------------------------------------
// y_mean / y_log_scale: hard-path gathers (out1/out2, [B,Ty,C]).
// ---------------------------------------------------------------------------
__global__ __launch_bounds__(160) void gather_ms_kernel(
    const float* __restrict__ o_mean, const float* __restrict__ o_ls,
    const int* __restrict__ IDX, float* __restrict__ out1,
    float* __restrict__ out2)
{
    int bs = blockIdx.x;            // b*TY_ + s
    int b  = bs >> 11;
    int tid = threadIdx.x;
    int id = IDX[bs];
    if (tid < C_) {
        int c = tid;
        out1[(size_t)bs*C_ + c] =
            (id >= 0) ? o_mean[(size_t)(b*C_ + c)*TX_ + id] : 0.f;
    } else {
        int c = tid - C_;
        out2[(size_t)bs*C_ + c] =
            (id >= 0) ? o_ls[(size_t)(b*C_ + c)*TX_ + id] : 0.f;
    }
}

// ---------------------------------------------------------------------------
// attn transposed [B,Ty,Tx]: streamed float4 one-hot write (67 MB).
// ---------------------------------------------------------------------------
__global__ __launch_bounds__(256) void attn_kernel(
    const int* __restrict__ IDX, float4* __restrict__ out3)
{
    int g = blockIdx.x * 256 + threadIdx.x;
    int e = g << 2;
    int t  = e & (TX_ - 1);
    int bs = e >> 9;                // b*TY_ + s
    int id = IDX[bs];
    float4 v;
    v.x = (t + 0 == id) ? 1.f : 0.f;
    v.y = (t + 1 == id) ? 1.f : 0.f;
    v.z = (t + 2 == id) ? 1.f : 0.f;
    v.w = (t + 3 == id) ? 1.f : 0.f;
    out3[g] = v;
}

extern "C" void kernel_launch(void* const* d_in, const int* in_sizes, int n_in,
                              void* d_out, int out_size, void* d_ws, size_t ws_size,
                              hipStream_t stream)
{
    const float* o_mean = (const float*)d_in[0];
    const float* o_ls   = (const float*)d_in[1];
    const float* z      = (const float*)d_in[2];
    const int*   xl     = (const int*)d_in[3];
    const int*   yl     = (const int*)d_in[4];

    char* ws = (char*)d_ws;
    float*    Apack = (float*)ws;    ws += (size_t)B_*TX_*K_*4;   //  5.0 MB
    float*    Bpack = (float*)ws;    ws += (size_t)B_*TY_*K_*4;   // 20.0 MB
    float*    S1    = (float*)ws;    ws += (size_t)B_*TX_*4;      // 32 KB
    float*    VAL   = (float*)ws;    ws += (size_t)B_*TY_*TX_*4;  // 64.0 MB
    unsigned* DEC   = (unsigned*)ws; ws += (size_t)B_*TY_*16*4;   //  2.0 MB
    int*      IDX   = (int*)ws;                                   // 128 KB

    float* out0 = (float*)d_out;                    // [B,Ty,C]
    float* out1 = out0 + (size_t)B_*TY_*C_;         // [B,Ty,C]
    float* out2 = out1 + (size_t)B_*TY_*C_;         // [B,Ty,C]
    float* out3 = out2 + (size_t)B_*TY_*C_;         // [B,Ty,Tx]
    float* out4 = out3 + (size_t)B_*TY_*TX_;        // [B,Tx,1]

    packA_kernel<<<dim3(TX_/32, B_), dim3(32, 8), 0, stream>>>(o_mean, o_ls, Apack, S1);
    packB_kernel<<<dim3(TY_/32, B_), dim3(32, 8), 0, stream>>>(z, yl, Bpack, out0);
    logp_wmma_kernel<<<dim3(TY_/16/8, TX_/16, B_), 256, 0, stream>>>(
        Apack, Bpack, S1, xl, yl, VAL);
    mas_forward_kernel<<<B_, 512, 0, stream>>>(VAL, DEC);
    mas_backward_kernel<<<B_, 256, 0, stream>>>(DEC, xl, yl, IDX, out4);
    gather_ms_kernel<<<B_*TY_, 160, 0, stream>>>(o_mean, o_ls, IDX, out1, out2);
    attn_kernel<<<(B_*TY_*TX_/4)/256, 256, 0, stream>>>(IDX, (float4*)out3);
}